// GCN_52484500357408
// MI455X (gfx1250) — compile-verified
//
#include <hip/hip_runtime.h>
#include <stdint.h>

// GCN 2-layer: out = GCNConv(relu(GCNConv(x, W1, b1)), W2, b2)
// Strategy:
//   scale[i]  = rsqrt(1 + indegree(i))                       (self-loop folded in)
//   g1        = scale * (x @ W1)          [WMMA f32 16x16x4]
//   acc1[i]   = g1[i]  (self-loop)  ;  acc1[dst] += g1[src]  (hw fp32 atomics)
//   acc1      = relu(scale*acc1 + b1)
//   g2        = scale * (acc1 @ W2)       [WMMA f32 16x16x4]
//   out[i]    = g2[i]  ;  out[dst] += g2[src]  ;  out = scale*out + b2

#define F0 20  // input features
#define F1 32  // hidden features
#define F2 20  // output features

typedef float v2f __attribute__((ext_vector_type(2)));
typedef float v8f __attribute__((ext_vector_type(8)));

__device__ __forceinline__ v8f wmma_f32x4(v2f a, v2f b, v8f c) {
  // V_WMMA_F32_16X16X4_F32 : D = A(16x4) x B(4x16) + C(16x16), all fp32
  return __builtin_amdgcn_wmma_f32_16x16x4_f32(
      /*neg_a=*/false, a, /*neg_b=*/false, b,
      /*c_mod=*/(short)0, c, /*reuse_a=*/false, /*reuse_b=*/false);
}

// ---------------- degree / normalization ----------------

__global__ void k_init_ones(float* __restrict__ p, unsigned n) {
  unsigned i = blockIdx.x * blockDim.x + threadIdx.x;
  if (i < n) p[i] = 1.0f;  // self-loop contributes 1 to every degree
}

__global__ void k_deg_accum(const long long* __restrict__ dst, float* __restrict__ deg,
                            unsigned E) {
  unsigned e = blockIdx.x * blockDim.x + threadIdx.x;
  if (e < E) unsafeAtomicAdd(&deg[(int)dst[e]], 1.0f);
}

__global__ void k_rsqrt(float* __restrict__ s, unsigned n) {
  unsigned i = blockIdx.x * blockDim.x + threadIdx.x;
  if (i < n) s[i] = rsqrtf(s[i]);  // deg >= 1 always (self-loops)
}

// ---------------- GEMM 1: g1 = scale * (x @ W1), [N,20]@[20,32] ----------------
// One wave computes a 16x32 output tile via two 16x16 WMMA accumulators.

__global__ void k_gemm1(const float* __restrict__ x, const float* __restrict__ W,
                        const float* __restrict__ scale, float* __restrict__ g,
                        unsigned N) {
  unsigned wave = (blockIdx.x * blockDim.x + threadIdx.x) >> 5;
  int lane = threadIdx.x & 31;
  int half = lane >> 4;        // lane half selects K-pair / M-upper-half
  int lr   = lane & 15;
  unsigned m0 = wave << 4;
  if (m0 >= N) return;         // wave-uniform exit: EXEC stays all-ones for WMMA

  v8f c0 = {0.f,0.f,0.f,0.f,0.f,0.f,0.f,0.f};
  v8f c1 = {0.f,0.f,0.f,0.f,0.f,0.f,0.f,0.f};
  const float* xr = x + (size_t)(m0 + lr) * F0;

#pragma unroll
  for (int k0 = 0; k0 < F0; k0 += 4) {
    int ka = k0 + 2 * half;                   // per-lane K index (ISA A/B layout)
    v2f a;  a.x = xr[ka];            a.y = xr[ka + 1];
    v2f b0; b0.x = W[ka * F1 + lr];  b0.y = W[(ka + 1) * F1 + lr];
    v2f b1; b1.x = W[ka * F1 + 16 + lr]; b1.y = W[(ka + 1) * F1 + 16 + lr];
    c0 = wmma_f32x4(a, b0, c0);
    c1 = wmma_f32x4(a, b1, c1);
  }

#pragma unroll
  for (int r = 0; r < 8; ++r) {
    unsigned row = m0 + r + 8 * half;
    float sc = scale[row];
    g[(size_t)row * F1 + lr]      = sc * c0[r];
    g[(size_t)row * F1 + 16 + lr] = sc * c1[r];
  }
}

// ---------------- GEMM 2: g2 = scale * (acc1 @ W2), [N,32]@[32,20] ----------------

__global__ void k_gemm2(const float* __restrict__ h, const float* __restrict__ W,
                        const float* __restrict__ scale, float* __restrict__ g,
                        unsigned N) {
  unsigned wave = (blockIdx.x * blockDim.x + threadIdx.x) >> 5;
  int lane = threadIdx.x & 31;
  int half = lane >> 4;
  int lr   = lane & 15;
  unsigned m0 = wave << 4;
  if (m0 >= N) return;

  v8f c0 = {0.f,0.f,0.f,0.f,0.f,0.f,0.f,0.f};
  v8f c1 = {0.f,0.f,0.f,0.f,0.f,0.f,0.f,0.f};
  const float* hr = h + (size_t)(m0 + lr) * F1;
  int colHi = 16 + lr;                 // columns 16..31, only <20 are real
  bool hiValid = (colHi < F2);

#pragma unroll
  for (int k0 = 0; k0 < F1; k0 += 4) {
    int ka = k0 + 2 * half;
    v2f a;  a.x = hr[ka];            a.y = hr[ka + 1];
    v2f b0; b0.x = W[ka * F2 + lr];  b0.y = W[(ka + 1) * F2 + lr];
    v2f b1;                           // zero-pad columns >= 20 (selects, not EXEC)
    b1.x = hiValid ? W[ka * F2 + colHi] : 0.0f;
    b1.y = hiValid ? W[(ka + 1) * F2 + colHi] : 0.0f;
    c0 = wmma_f32x4(a, b0, c0);
    c1 = wmma_f32x4(a, b1, c1);
  }

#pragma unroll
  for (int r = 0; r < 8; ++r) {
    unsigned row = m0 + r + 8 * half;
    float sc = scale[row];
    g[(size_t)row * F2 + lr] = sc * c0[r];
    if (hiValid) g[(size_t)row * F2 + colHi] = sc * c1[r];
  }
}

// ---------------- copies / scatters / epilogues ----------------

__global__ void k_copy(const float* __restrict__ src, float* __restrict__ dst, unsigned n) {
  unsigned i = blockIdx.x * blockDim.x + threadIdx.x;
  if (i < n) dst[i] = src[i];
}

// layer-1 scatter: one lane per (edge, feature), 32 features -> one wave per edge
__global__ void k_scatter32(const long long* __restrict__ src, const long long* __restrict__ dst,
                            const float* __restrict__ g, float* __restrict__ acc, unsigned E) {
  unsigned gid = blockIdx.x * blockDim.x + threadIdx.x;
  unsigned e = gid >> 5;
  unsigned f = gid & 31;
  if (e < E) {
    int s = (int)src[e];
    int d = (int)dst[e];
    unsafeAtomicAdd(&acc[(size_t)d * F1 + f], g[(size_t)s * F1 + f]);
  }
}

// layer-2 scatter: one lane per (edge, feature), 20 features
__global__ void k_scatter20(const long long* __restrict__ src, const long long* __restrict__ dst,
                            const float* __restrict__ g, float* __restrict__ acc, unsigned total,
                            unsigned /*E*/) {
  unsigned gid = blockIdx.x * blockDim.x + threadIdx.x;
  if (gid < total) {
    unsigned e = gid / F2;
    unsigned f = gid - e * F2;
    int s = (int)src[e];
    int d = (int)dst[e];
    unsafeAtomicAdd(&acc[(size_t)d * F2 + f], g[(size_t)s * F2 + f]);
  }
}

__global__ void k_post1(float* __restrict__ acc, const float* __restrict__ scale,
                        const float* __restrict__ b, unsigned n) {
  unsigned j = blockIdx.x * blockDim.x + threadIdx.x;
  if (j < n) {
    unsigned i = j >> 5;        // F1 == 32
    unsigned f = j & 31;
    acc[j] = fmaxf(fmaf(scale[i], acc[j], b[f]), 0.0f);
  }
}

__global__ void k_final(float* __restrict__ out, const float* __restrict__ scale,
                        const float* __restrict__ b, unsigned n) {
  unsigned j = blockIdx.x * blockDim.x + threadIdx.x;
  if (j < n) {
    unsigned i = j / F2;
    unsigned f = j - i * F2;
    out[j] = fmaf(scale[i], out[j], b[f]);
  }
}

// ---------------- host side ----------------

static inline unsigned cdiv(unsigned long long a, unsigned b) {
  return (unsigned)((a + b - 1) / b);
}

extern "C" void kernel_launch(void* const* d_in, const int* in_sizes, int n_in,
                              void* d_out, int out_size, void* d_ws, size_t ws_size,
                              hipStream_t stream) {
  const float*     x   = (const float*)d_in[0];
  const long long* ei  = (const long long*)d_in[1];   // int64 [2, E]
  const float*     W1  = (const float*)d_in[2];
  const float*     b1  = (const float*)d_in[3];
  const float*     W2  = (const float*)d_in[4];
  const float*     b2  = (const float*)d_in[5];
  float*           out = (float*)d_out;

  const unsigned N = (unsigned)(in_sizes[0] / F0);
  const unsigned E = (unsigned)(in_sizes[1] / 2);
  const long long* srcIdx = ei;        // edge_index[0]
  const long long* dstIdx = ei + E;    // edge_index[1]

  // workspace layout (floats)
  float* ws    = (float*)d_ws;
  size_t Na    = ((size_t)N + 63) & ~(size_t)63;
  float* scale = ws;                    // [N]    degree -> rsqrt in place
  float* g1    = scale + Na;            // [N,32] pre-scaled layer-1 features
  float* acc1  = g1 + (size_t)N * F1;   // [N,32] layer-1 accumulator / relu output
  float* g2    = acc1 + (size_t)N * F1; // [N,20] pre-scaled layer-2 features
  (void)ws_size; (void)n_in; (void)out_size;

  const unsigned B = 256;

  // normalization
  k_init_ones<<<cdiv(N, B), B, 0, stream>>>(scale, N);
  k_deg_accum<<<cdiv(E, B), B, 0, stream>>>(dstIdx, scale, E);
  k_rsqrt<<<cdiv(N, B), B, 0, stream>>>(scale, N);

  // layer 1
  unsigned waves1 = (N + 15) / 16;
  k_gemm1<<<cdiv((unsigned long long)waves1 * 32, B), B, 0, stream>>>(x, W1, scale, g1, N);
  k_copy<<<cdiv((unsigned long long)N * F1, B), B, 0, stream>>>(g1, acc1, N * F1);
  k_scatter32<<<cdiv((unsigned long long)E * 32, B), B, 0, stream>>>(srcIdx, dstIdx, g1, acc1, E);
  k_post1<<<cdiv((unsigned long long)N * F1, B), B, 0, stream>>>(acc1, scale, b1, N * F1);

  // layer 2
  k_gemm2<<<cdiv((unsigned long long)waves1 * 32, B), B, 0, stream>>>(acc1, W2, scale, g2, N);
  k_copy<<<cdiv((unsigned long long)N * F2, B), B, 0, stream>>>(g2, out, N * F2);
  unsigned tot2 = E * F2;
  k_scatter20<<<cdiv(tot2, B), B, 0, stream>>>(srcIdx, dstIdx, g2, out, tot2, E);
  k_final<<<cdiv((unsigned long long)N * F2, B), B, 0, stream>>>(out, scale, b2, N * F2);
}